// DAGSCM_82729660055621
// MI455X (gfx1250) — compile-verified
//
#include <hip/hip_runtime.h>
#include <stdint.h>

#define N_NODES 128
#define N_OBS   64
#define TILE_STRIDE 33   // 32 samples + 1 pad dword per node row (bank-conflict-free transpose)

#if defined(__has_builtin)
#  if __has_builtin(__builtin_amdgcn_tensor_load_to_lds) && __has_builtin(__builtin_amdgcn_s_wait_tensorcnt)
#    define HAVE_TDM 1
#  endif
#  if __has_builtin(__builtin_amdgcn_global_load_async_to_lds_b32)
#    define HAVE_ASYNC 1
#  endif
#endif

// ---- compile-time probe: which staging path did the preprocessor select? ----
#if defined(HAVE_TDM)
#  if __clang_major__ >= 23
#    warning "CDNA5 staging path: TDM tensor_load_to_lds (6-arg form, clang>=23)"
#  else
#    warning "CDNA5 staging path: TDM tensor_load_to_lds (5-arg form, clang<23)"
#  endif
#elif defined(HAVE_ASYNC)
#  warning "CDNA5 staging path: global_load_async_to_lds_b32 + s_wait_asynccnt"
#else
#  warning "CDNA5 staging path: FALLBACK plain global->LDS loads (no async/tensor builtins found)"
#endif

typedef unsigned int uv4 __attribute__((ext_vector_type(4)));
typedef int          iv4 __attribute__((ext_vector_type(4)));
typedef int          iv8 __attribute__((ext_vector_type(8)));

#define AS1 __attribute__((address_space(1)))
#define AS3 __attribute__((address_space(3)))

// ---------------- ws layout ----------------
// [0,2048)      float4[128]  packed params {w0*m0, w1*m1, b, sigma}
// [2048,2560)   uint[128]    meta  {p0 | p1<<8 | root<<16}
// [4096, +128*cal_n*4)       pilot values [node][cal_sample]

// K3 (defined FIRST so the disasm snippet shows the TDM/async staging):
// main pass. 2 waves/block; each wave owns one 128x32 tile (stride 33) staged by TDM.
__global__ void __launch_bounds__(64)
k3_main(const float4* __restrict__ par, const unsigned* __restrict__ meta,
        const float* __restrict__ root_main, const float* __restrict__ z_noise,
        const int* __restrict__ chosen, float* __restrict__ out,
        unsigned long long ns)
{
    __shared__ float4   sPar[N_NODES];
    __shared__ unsigned sMeta[N_NODES];
    __shared__ float    sTile[2][N_NODES * TILE_STRIDE];

    const int tid  = threadIdx.x;
    const int wid  = tid >> 5;
    const int lane = tid & 31;
    const unsigned long long base = ((unsigned long long)blockIdx.x * 2ull + (unsigned)wid) * 32ull;
    float* tile = &sTile[wid][0];

    // ---- stage this wave's z tile: 128 rows x 32 samples, padded rows (stride 33) ----
#if defined(HAVE_TDM)
    {
        unsigned ldsAddr = (unsigned)(unsigned long long)(uintptr_t)tile;
        unsigned long long ga = (unsigned long long)(uintptr_t)z_noise + base * 4ull;
        uv4 g0;
        g0[0] = 1u;                                          // count=1, user-mode load
        g0[1] = ldsAddr;                                     // lds_addr (bytes)
        g0[2] = (unsigned)(ga & 0xFFFFFFFFull);              // global_addr[31:0]
        g0[3] = (unsigned)((ga >> 32) & 0x1FFFFFFull) | (2u << 30);  // addr[56:32] | type=2
        iv8 g1;
        // data_size=4B(2), pad_enable, pad_interval=4 (32 dwords/row), pad_amount=0 (1 dword)
        g1[0] = (int)((2u << 16) | (1u << 20) | (4u << 22));
        g1[1] = (int)(((unsigned)(ns & 0xFFFFull)) << 16);                         // tensor_dim0[15:0]
        g1[2] = (int)(((unsigned)((ns >> 16) & 0xFFFFull)) | (128u << 16));        // dim0[31:16] | tensor_dim1=128
        g1[3] = (int)(32u << 16);                                                  // dim1 hi=0 | tile_dim0=32
        g1[4] = (int)128u;                                                         // tile_dim1=128, tile_dim2=0
        g1[5] = (int)(unsigned)(ns & 0xFFFFFFFFull);                               // tensor_dim0_stride lo32
        g1[6] = (int)(unsigned)((ns >> 32) & 0xFFFFull);                           // stride hi16 | dim1_stride lo=0
        g1[7] = 0;
        iv4 gz = {0, 0, 0, 0};
#if __clang_major__ >= 23
        iv8 gz8 = {0, 0, 0, 0, 0, 0, 0, 0};
        __builtin_amdgcn_tensor_load_to_lds(g0, g1, gz, gz, gz8, 0);
#else
        __builtin_amdgcn_tensor_load_to_lds(g0, g1, gz, gz, 0);
#endif
    }
#elif defined(HAVE_ASYNC)
    for (int r = 0; r < N_NODES; ++r) {
        const float* gp = z_noise + (unsigned long long)r * ns + base + (unsigned)lane;
        float* lp = tile + r * TILE_STRIDE + lane;
        __builtin_amdgcn_global_load_async_to_lds_b32(
            (AS1 void*)(uintptr_t)gp, (AS3 void*)(uintptr_t)lp, 0, 0);
    }
#else
    for (int r = 0; r < N_NODES; ++r)
        tile[r * TILE_STRIDE + lane] = z_noise[(unsigned long long)r * ns + base + (unsigned)lane];
#endif

    // ---- stage packed node params (shared by both waves) while the DMA runs ----
    for (int k = tid; k < N_NODES; k += 64) { sPar[k] = par[k]; sMeta[k] = meta[k]; }
    __syncthreads();

#if defined(HAVE_TDM)
    __builtin_amdgcn_s_wait_tensorcnt(0);
#elif defined(HAVE_ASYNC)
#  if __has_builtin(__builtin_amdgcn_s_wait_asynccnt)
    __builtin_amdgcn_s_wait_asynccnt(0);
#  else
    asm volatile("s_wait_asynccnt 0" ::: "memory");
#  endif
#endif

    // ---- topological scan: z row is consumed, node value written back in place ----
    #pragma unroll 4
    for (int i = 0; i < N_NODES; ++i) {
        float4 P = sPar[i];
        unsigned m = (unsigned)__builtin_amdgcn_readfirstlane((int)sMeta[i]);
        float zv = tile[i * TILE_STRIDE + lane];
        float v;
        if (m & (1u << 16)) {   // root: take root_main row (noise not applied to roots)
            v = root_main[(unsigned long long)i * ns + base + (unsigned)lane];
        } else {
            float v0 = tile[(m & 255u) * TILE_STRIDE + lane];
            float v1 = tile[((m >> 8) & 255u) * TILE_STRIDE + lane];
            v = fmaxf(fmaf(P.x, v0, fmaf(P.y, v1, P.z)), 0.0f) + P.w * zv;
        }
        v = (fabsf(v) < __builtin_inff()) ? v : 0.0f;   // isfinite clamp
        tile[i * TILE_STRIDE + lane] = v;
    }

    // ---- gather chosen rows + transpose: coalesced b128 stores (512B per instruction) ----
    const int4 cv = ((const int4*)chosen)[lane & 15];   // obs columns 4*(lane&15)..+3
    const int shalf = lane >> 4;                        // lanes 0-15: sample 2j, 16-31: 2j+1
    #pragma unroll 4
    for (int j = 0; j < 16; ++j) {
        int sl = 2 * j + shalf;
        float4 o;
        o.x = tile[cv.x * TILE_STRIDE + sl];
        o.y = tile[cv.y * TILE_STRIDE + sl];
        o.z = tile[cv.z * TILE_STRIDE + sl];
        o.w = tile[cv.w * TILE_STRIDE + sl];
        float4* op = (float4*)(out + (base + (unsigned long long)sl) * N_OBS);
        op[lane & 15] = o;
    }
}

// K0: pack per-node parameters. Root <=> both parent masks zero (matches DAG construction).
__global__ void __launch_bounds__(128)
k0_pack(const float* __restrict__ W, const float* __restrict__ b,
        const float* __restrict__ mask, const int* __restrict__ pidx,
        float4* __restrict__ par, unsigned* __restrict__ meta)
{
    int i = threadIdx.x;
    if (i >= N_NODES) return;
    float m0 = mask[2*i], m1 = mask[2*i+1];
    float w0 = W[2*i] * m0, w1 = W[2*i+1] * m1;
    unsigned p0 = (unsigned)pidx[2*i]   & 255u;
    unsigned p1 = (unsigned)pidx[2*i+1] & 255u;
    unsigned root = (m0 == 0.0f && m1 == 0.0f) ? 1u : 0u;
    par[i]  = make_float4(w0, w1, b[i], 0.0f);
    meta[i] = p0 | (p1 << 8) | (root << 16);
}

// K1: noiseless pilot scan; one wave handles 32 pilot samples, vals live in LDS.
__global__ void __launch_bounds__(32)
k1_pilot(const float4* __restrict__ par, const unsigned* __restrict__ meta,
         const float* __restrict__ root_pilot, float* __restrict__ pilot_out, int cal_n)
{
    __shared__ float4   sPar[N_NODES];
    __shared__ unsigned sMeta[N_NODES];
    __shared__ float    sV[N_NODES * 32];
    const int lane = threadIdx.x;
    const int qs   = blockIdx.x * 32;
    for (int k = lane; k < N_NODES; k += 32) { sPar[k] = par[k]; sMeta[k] = meta[k]; }
    __syncthreads();
    for (int i = 0; i < N_NODES; ++i) {
        float4 P = sPar[i];
        unsigned m = (unsigned)__builtin_amdgcn_readfirstlane((int)sMeta[i]);
        float v;
        if (m & (1u << 16)) {
            v = root_pilot[(size_t)i * cal_n + qs + lane];
        } else {
            float v0 = sV[(m & 255u) * 32 + lane];
            float v1 = sV[((m >> 8) & 255u) * 32 + lane];
            v = fmaxf(fmaf(P.x, v0, fmaf(P.y, v1, P.z)), 0.0f);
        }
        v = (fabsf(v) < __builtin_inff()) ? v : 0.0f;   // isfinite clamp
        sV[i * 32 + lane] = v;
        pilot_out[(size_t)i * cal_n + qs + lane] = v;   // coalesced row store
    }
}

// K2: exact order statistics via rank counting; jnp.quantile 'linear' interpolation.
__global__ void __launch_bounds__(256)
k2_sigma(const float* __restrict__ pilot, float4* __restrict__ par, int n)
{
    __shared__ float a[256];
    __shared__ float q[4];
    const int i = blockIdx.x;
    const int t = threadIdx.x;
    float x = 0.0f;
    if (t < n) x = pilot[(size_t)i * n + t];
    a[t] = x;
    __syncthreads();
    const float p25 = 0.25f * (n - 1), p75 = 0.75f * (n - 1);
    const int l25 = (int)p25, l75 = (int)p75;
    const int h25 = (l25 + 1 < n) ? l25 + 1 : l25;
    const int h75 = (l75 + 1 < n) ? l75 + 1 : l75;
    if (t < n) {
        int r = 0;
        for (int u = 0; u < n; ++u) {
            float au = a[u];
            r += (au < x) || (au == x && u < t);   // exact rank (tie-break by index)
        }
        if (r == l25) q[0] = x;
        if (r == h25) q[1] = x;
        if (r == l75) q[2] = x;
        if (r == h75) q[3] = x;
    }
    __syncthreads();
    if (t == 0) {
        float f25 = p25 - (float)l25, f75 = p75 - (float)l75;
        float q25 = q[0] + f25 * (q[1] - q[0]);
        float q75 = q[2] + f75 * (q[3] - q[2]);
        par[i].w = 0.1f * fmaxf(q75 - q25, 1e-6f);   // patch sigma into packed params
    }
}

extern "C" void kernel_launch(void* const* d_in, const int* in_sizes, int n_in,
                              void* d_out, int out_size, void* d_ws, size_t ws_size,
                              hipStream_t stream)
{
    // setup_inputs order: n_samples, W, b, root_pilot, root_main, z_noise,
    //                     par_mask, par_idx, is_root, chosen
    const float* W      = (const float*)d_in[1];
    const float* b      = (const float*)d_in[2];
    const float* pilotR = (const float*)d_in[3];
    const float* rootM  = (const float*)d_in[4];
    const float* zN     = (const float*)d_in[5];
    const float* mask   = (const float*)d_in[6];
    const int*   pidx   = (const int*)d_in[7];
    const int*   chosen = (const int*)d_in[9];
    float* out = (float*)d_out;

    unsigned long long ns = (unsigned long long)in_sizes[4] / N_NODES;  // 262144
    int cal_n = in_sizes[3] / N_NODES;                                  // 256

    char* ws = (char*)d_ws;
    float4*   wsPar   = (float4*)(ws + 0);
    unsigned* wsMeta  = (unsigned*)(ws + 2048);
    float*    wsPilot = (float*)(ws + 4096);

    k0_pack <<<1, 128, 0, stream>>>(W, b, mask, pidx, wsPar, wsMeta);
    k1_pilot<<<cal_n / 32, 32, 0, stream>>>(wsPar, wsMeta, pilotR, wsPilot, cal_n);
    k2_sigma<<<N_NODES, 256, 0, stream>>>(wsPilot, wsPar, cal_n);
    k3_main <<<(unsigned)(ns / 64), 64, 0, stream>>>(wsPar, wsMeta, rootM, zN, chosen, out, ns);
}